// SOMLayer_20504173871532
// MI455X (gfx1250) — compile-verified
//
#include <hip/hip_runtime.h>
#include <hip/hip_bf16.h>
#include <math.h>

typedef __attribute__((ext_vector_type(2))) float v2f;
typedef __attribute__((ext_vector_type(8))) float v8f;
typedef int v4i __attribute__((vector_size(16)));

#define AS1 __attribute__((address_space(1)))
#define AS3 __attribute__((address_space(3)))

#define BATCH     8192
#define G_TOTAL   4096
#define DIM       1024
#define GW        64
#define LR        0.1f

#if defined(__has_builtin)
#if __has_builtin(__builtin_amdgcn_global_load_async_to_lds_b128)
#define HAVE_ASYNC_LDS 1
#endif
#endif

#if defined(__has_builtin)
#if __has_builtin(__builtin_amdgcn_s_wait_asynccnt)
#define WAIT_ASYNC(n) __builtin_amdgcn_s_wait_asynccnt(n)
#endif
#endif
#ifndef WAIT_ASYNC
#define WAIT_ASYNC(n) asm volatile("s_wait_asynccnt %0" ::"i"(n) : "memory")
#endif

// ---------------------------------------------------------------------------
// ws layout (bytes)
// ---------------------------------------------------------------------------
#define WS_W2     (0)                       // float[4096]
#define WS_X2     (16*1024)                 // float[8192]
#define WS_KEYS   (48*1024)                 // u64[8192]
#define WS_BMU    (112*1024)                // int[8192]
#define WS_CNT    (144*1024)                // float[4096]
#define WS_SCOL   (160*1024)                // float[64*64]
#define WS_XSUM   (176*1024)                // float[4096*1024]  (16 MB)
#define WS_TMP    (176*1024 + 16*1024*1024) // float[4096*1024]  (16 MB)

// ---------------------------------------------------------------------------
// 0) init: zero Xsum + cnt, saturate argmin keys
// ---------------------------------------------------------------------------
__global__ void som_init(float* __restrict__ Xsum, float* __restrict__ cnt,
                         unsigned long long* __restrict__ keys) {
    int idx = blockIdx.x * 256 + threadIdx.x;
    if (idx < G_TOTAL * DIM) Xsum[idx] = 0.0f;
    if (idx < BATCH)         keys[idx] = 0xFFFFFFFFFFFFFFFFull;
    if (idx < G_TOTAL)       cnt[idx]  = 0.0f;
}

// ---------------------------------------------------------------------------
// 1) squared row norms: rows [0,8192) -> x2, rows [8192,12288) -> w2
// ---------------------------------------------------------------------------
__global__ __launch_bounds__(256) void som_sqnorm(const float* __restrict__ X,
                                                  const float* __restrict__ W,
                                                  float* __restrict__ x2,
                                                  float* __restrict__ w2) {
    const int row = blockIdx.x;
    const float* src;
    float* dst;
    if (row < BATCH) { src = X + (size_t)row * DIM;           dst = x2 + row; }
    else             { src = W + (size_t)(row - BATCH) * DIM; dst = w2 + (row - BATCH); }
    float s = 0.0f;
    for (int i = threadIdx.x; i < DIM; i += 256) { float v = src[i]; s += v * v; }
    __shared__ float red[256];
    red[threadIdx.x] = s;
    __syncthreads();
    for (int off = 128; off > 0; off >>= 1) {
        if (threadIdx.x < off) red[threadIdx.x] += red[threadIdx.x + off];
        __syncthreads();
    }
    if (threadIdx.x == 0) *dst = red[0];
}

// ---------------------------------------------------------------------------
// 2) distance GEMM + argmin.  fp32 WMMA 16x16x4.
//    Workgroup tile: 128 (batch) x 64 (neurons); 8 waves as 4x2,
//    each wave owns a 32x32 subtile = 2x2 WMMA accumulators.
//    Double-buffered LDS staged via GLOBAL_LOAD_ASYNC_TO_LDS_B128 (ASYNCcnt).
// ---------------------------------------------------------------------------
#define BM 128
#define BN 64
#define KC 16
#define LDP 20  // padded LDS row stride (floats): conflict-free & float4-aligned

__device__ inline unsigned long long shfl_xor_u64_16(unsigned long long v, int m) {
    unsigned int lo = (unsigned int)v;
    unsigned int hi = (unsigned int)(v >> 32);
    lo = __shfl_xor(lo, m, 16);
    hi = __shfl_xor(hi, m, 16);
    return ((unsigned long long)hi << 32) | lo;
}

#ifdef HAVE_ASYNC_LDS
__device__ __forceinline__ void stage_async(const float* __restrict__ X,
                                            const float* __restrict__ W,
                                            float* XsB, float* WsB,
                                            int m0, int n0, int kc, int tid) {
    // X tile: 128x16 floats = 512 b128 transfers, 2 per thread
    #pragma unroll
    for (int p = 0; p < 2; ++p) {
        const int idx = tid + p * 256;
        const int row = idx >> 2, c4 = idx & 3;
        __builtin_amdgcn_global_load_async_to_lds_b128(
            (AS1 v4i*)(unsigned long long)(const void*)(X + (size_t)(m0 + row) * DIM + kc + c4 * 4),
            (AS3 v4i*)(XsB + row * LDP + c4 * 4), 0, 0);
    }
    // W tile: 64x16 floats = 256 b128 transfers, 1 per thread
    const int row = tid >> 2, c4 = tid & 3;
    __builtin_amdgcn_global_load_async_to_lds_b128(
        (AS1 v4i*)(unsigned long long)(const void*)(W + (size_t)(n0 + row) * DIM + kc + c4 * 4),
        (AS3 v4i*)(WsB + row * LDP + c4 * 4), 0, 0);
}
#endif

__global__ __launch_bounds__(256) void som_dist(const float* __restrict__ X,
                                                const float* __restrict__ W,
                                                const float* __restrict__ x2,
                                                const float* __restrict__ w2,
                                                unsigned long long* __restrict__ keys) {
    __shared__ __align__(16) float Xs[2][BM * LDP];
    __shared__ __align__(16) float Ws[2][BN * LDP];

    const int m0   = blockIdx.x * BM;
    const int n0   = blockIdx.y * BN;
    const int tid  = threadIdx.x;
    const int lane = tid & 31;
    const int wave = tid >> 5;
    const int wm   = wave >> 1;   // 0..3
    const int wn   = wave & 1;    // 0..1
    const int lo   = lane & 15;
    const int hi   = lane >> 4;   // 0 or 1

    v8f c[2][2] = {};

#ifdef HAVE_ASYNC_LDS
    // ---------------- async double-buffered pipeline ----------------
    stage_async(X, W, Xs[0], Ws[0], m0, n0, 0, tid);
    for (int kc = 0, it = 0; kc < DIM; kc += KC, ++it) {
        const int buf = it & 1;
        __syncthreads();  // all waves done reading buf^1 -> safe to overwrite
        if (kc + KC < DIM) {
            stage_async(X, W, Xs[buf ^ 1], Ws[buf ^ 1], m0, n0, kc + KC, tid);
            WAIT_ASYNC(3);   // chunk kc complete; next chunk (3 ops) in flight
        } else {
            WAIT_ASYNC(0);
        }
        __syncthreads();     // chunk kc visible to every wave

        const float* Xb = Xs[buf];
        const float* Wb = Ws[buf];
        #pragma unroll
        for (int k4 = 0; k4 < KC; k4 += 4) {
            v2f a[2], b[2];
            #pragma unroll
            for (int sm = 0; sm < 2; ++sm) {
                const int m = wm * 32 + sm * 16 + lo;
                a[sm].x = Xb[m * LDP + k4 + 2 * hi];
                a[sm].y = Xb[m * LDP + k4 + 2 * hi + 1];
            }
            #pragma unroll
            for (int sn = 0; sn < 2; ++sn) {
                const int n = wn * 32 + sn * 16 + lo;
                b[sn].x = Wb[n * LDP + k4 + 2 * hi];
                b[sn].y = Wb[n * LDP + k4 + 2 * hi + 1];
            }
            c[0][0] = __builtin_amdgcn_wmma_f32_16x16x4_f32(false, a[0], false, b[0], (short)0, c[0][0], false, false);
            c[0][1] = __builtin_amdgcn_wmma_f32_16x16x4_f32(false, a[0], false, b[1], (short)0, c[0][1], false, false);
            c[1][0] = __builtin_amdgcn_wmma_f32_16x16x4_f32(false, a[1], false, b[0], (short)0, c[1][0], false, false);
            c[1][1] = __builtin_amdgcn_wmma_f32_16x16x4_f32(false, a[1], false, b[1], (short)0, c[1][1], false, false);
        }
    }
#else
    // ---------------- fallback: synchronous staging ----------------
    for (int kc = 0; kc < DIM; kc += KC) {
        __syncthreads();
        {
            #pragma unroll
            for (int p = 0; p < 2; ++p) {
                int idx = tid + p * 256;
                int row = idx >> 2, c4 = idx & 3;
                float4 v = *(const float4*)(X + (size_t)(m0 + row) * DIM + kc + c4 * 4);
                *(float4*)(&Xs[0][row * LDP + c4 * 4]) = v;
            }
            int row = tid >> 2, c4 = tid & 3;
            float4 v = *(const float4*)(W + (size_t)(n0 + row) * DIM + kc + c4 * 4);
            *(float4*)(&Ws[0][row * LDP + c4 * 4]) = v;
        }
        if (kc + KC < DIM) {
            int row = tid >> 1;
            __builtin_prefetch(X + (size_t)(m0 + row) * DIM + kc + KC, 0, 3);
            if (tid < 128)
                __builtin_prefetch(W + (size_t)(n0 + (tid >> 1)) * DIM + kc + KC, 0, 3);
        }
        __syncthreads();

        #pragma unroll
        for (int k4 = 0; k4 < KC; k4 += 4) {
            v2f a[2], b[2];
            #pragma unroll
            for (int sm = 0; sm < 2; ++sm) {
                const int m = wm * 32 + sm * 16 + lo;
                a[sm].x = Xs[0][m * LDP + k4 + 2 * hi];
                a[sm].y = Xs[0][m * LDP + k4 + 2 * hi + 1];
            }
            #pragma unroll
            for (int sn = 0; sn < 2; ++sn) {
                const int n = wn * 32 + sn * 16 + lo;
                b[sn].x = Ws[0][n * LDP + k4 + 2 * hi];
                b[sn].y = Ws[0][n * LDP + k4 + 2 * hi + 1];
            }
            c[0][0] = __builtin_amdgcn_wmma_f32_16x16x4_f32(false, a[0], false, b[0], (short)0, c[0][0], false, false);
            c[0][1] = __builtin_amdgcn_wmma_f32_16x16x4_f32(false, a[0], false, b[1], (short)0, c[0][1], false, false);
            c[1][0] = __builtin_amdgcn_wmma_f32_16x16x4_f32(false, a[1], false, b[0], (short)0, c[1][0], false, false);
            c[1][1] = __builtin_amdgcn_wmma_f32_16x16x4_f32(false, a[1], false, b[1], (short)0, c[1][1], false, false);
        }
    }
#endif

    // epilogue: dist = x2 + w2 - 2*dot, clamp, pack (dist_bits, g), argmin
    #pragma unroll
    for (int sm = 0; sm < 2; ++sm) {
        #pragma unroll
        for (int r = 0; r < 8; ++r) {
            const int mg = m0 + wm * 32 + sm * 16 + 8 * hi + r;
            const float xx = x2[mg];
            unsigned long long best = 0xFFFFFFFFFFFFFFFFull;
            #pragma unroll
            for (int sn = 0; sn < 2; ++sn) {
                const int ng = n0 + wn * 32 + sn * 16 + lo;
                float d = xx + w2[ng] - 2.0f * c[sm][sn][r];
                d = fmaxf(d, 0.0f);
                unsigned long long key =
                    ((unsigned long long)__float_as_uint(d) << 32) | (unsigned int)ng;
                best = key < best ? key : best;
            }
            #pragma unroll
            for (int off = 8; off > 0; off >>= 1) {
                unsigned long long o = shfl_xor_u64_16(best, off);
                best = o < best ? o : best;
            }
            if (lo == 0) atomicMin(keys + mg, best);  // global_atomic_min_b64
        }
    }
}

// ---------------------------------------------------------------------------
// 3) decode BMU keys -> outputs (indices, qerr), bmu list, BMU histogram
// ---------------------------------------------------------------------------
__global__ void som_bmu(const unsigned long long* __restrict__ keys,
                        int* __restrict__ bmu, float* __restrict__ cnt,
                        float* __restrict__ out) {
    int b = blockIdx.x * 256 + threadIdx.x;
    if (b >= BATCH) return;
    unsigned long long k = keys[b];
    unsigned int g = (unsigned int)(k & 0xFFFFFFFFu);
    float d = __uint_as_float((unsigned int)(k >> 32));
    out[2 * b]     = (float)(g >> 6);   // bmu row
    out[2 * b + 1] = (float)(g & 63);   // bmu col
    out[2 * BATCH + b] = sqrtf(d);      // quantization error
    bmu[b] = (int)g;
    atomicAdd(cnt + g, 1.0f);
}

// ---------------------------------------------------------------------------
// 4) scatter-add sample rows into per-BMU-cell sums
// ---------------------------------------------------------------------------
__global__ __launch_bounds__(256) void som_scatter(const float* __restrict__ X,
                                                   const int* __restrict__ bmu,
                                                   float* __restrict__ Xsum) {
    const int b = blockIdx.x;
    const int g = bmu[b];
    const float* xr = X + (size_t)b * DIM;
    float* dst = Xsum + (size_t)g * DIM;
    for (int d = threadIdx.x; d < DIM; d += 256)
        atomicAdd(dst + d, xr[d]);       // global_atomic_add_f32
}

// ---------------------------------------------------------------------------
// 5) separable Gaussian transform, stage 1 (columns):
//    tmp[rb][gc][d] = sum_cb E[cb][gc] * Xsum[rb*64+cb][d]
//    Scol[rb][gc]   = sum_cb E[cb][gc] * cnt[rb*64+cb]
// ---------------------------------------------------------------------------
__global__ __launch_bounds__(128) void som_trans1(const float* __restrict__ Xsum,
                                                  const float* __restrict__ cnt,
                                                  float* __restrict__ tmp,
                                                  float* __restrict__ Scol) {
    __shared__ float E[64 * 64];
    __shared__ float Xls[64 * 128];
    const int rb  = blockIdx.x;
    const int d0  = blockIdx.y * 128;
    const int tid = threadIdx.x;

    for (int idx = tid; idx < 64 * 64; idx += 128) {
        float df = (float)((idx >> 6) - (idx & 63));
        E[idx] = expf(-0.5f * df * df);
    }
    for (int cb = 0; cb < 64; ++cb)
        Xls[cb * 128 + tid] = Xsum[(size_t)((rb * 64 + cb) << 10) + d0 + tid];
    __syncthreads();

    if (blockIdx.y == 0 && tid < 64) {
        float s = 0.0f;
        for (int cb = 0; cb < 64; ++cb) s += E[cb * 64 + tid] * cnt[rb * 64 + cb];
        Scol[rb * 64 + tid] = s;
    }

    for (int gc = 0; gc < 64; ++gc) {
        float acc = 0.0f;
        #pragma unroll
        for (int cb = 0; cb < 64; ++cb)
            acc += E[cb * 64 + gc] * Xls[cb * 128 + tid];
        tmp[(size_t)((rb * 64 + gc) << 10) + d0 + tid] = acc;
    }
}

// ---------------------------------------------------------------------------
// 6) stage 2 (rows) fused with the weight update:
//    acc[g][d] = sum_rb E[rb][gr] * tmp[rb][gc][d];  S likewise from Scol
//    new_w = w + LR*(acc - S*w)
// ---------------------------------------------------------------------------
__global__ __launch_bounds__(256) void som_update(const float* __restrict__ W,
                                                  const float* __restrict__ tmp,
                                                  const float* __restrict__ Scol,
                                                  float* __restrict__ outw) {
    __shared__ float El[64];
    __shared__ float Ssh;
    const int g   = blockIdx.x;
    const int gr  = g >> 6;
    const int gc  = g & 63;
    const int tid = threadIdx.x;

    if (tid < 64) {
        float df = (float)(tid - gr);
        El[tid] = expf(-0.5f * df * df);
    }
    __syncthreads();
    if (tid == 0) {
        float s = 0.0f;
        for (int rb = 0; rb < 64; ++rb) s += El[rb] * Scol[rb * 64 + gc];
        Ssh = s;
    }
    __syncthreads();
    const float S = Ssh;

    for (int j = 0; j < DIM / 256; ++j) {
        const int d = tid + j * 256;
        float acc = 0.0f;
        for (int rb = 0; rb < 64; ++rb)
            acc += El[rb] * tmp[(size_t)((rb * 64 + gc) << 10) + d];
        const float w = W[(size_t)(g << 10) + d];
        outw[(size_t)(g << 10) + d] = w + LR * (acc - S * w);
    }
}

// ---------------------------------------------------------------------------
extern "C" void kernel_launch(void* const* d_in, const int* in_sizes, int n_in,
                              void* d_out, int out_size, void* d_ws, size_t ws_size,
                              hipStream_t stream) {
    const float* X  = (const float*)d_in[0];   // [8192, 1024]
    const float* Wm = (const float*)d_in[1];   // [64, 64, 1024]
    float* out = (float*)d_out;                // [16384 idx | 8192 qerr | 4194304 weights]
    char* ws = (char*)d_ws;

    float* w2  = (float*)(ws + WS_W2);
    float* x2  = (float*)(ws + WS_X2);
    unsigned long long* keys = (unsigned long long*)(ws + WS_KEYS);
    int*   bmu  = (int*)(ws + WS_BMU);
    float* cnt  = (float*)(ws + WS_CNT);
    float* Scol = (float*)(ws + WS_SCOL);
    float* Xsum = (float*)(ws + WS_XSUM);
    float* tmp  = (float*)(ws + WS_TMP);

    som_init<<<(G_TOTAL * DIM + 255) / 256, 256, 0, stream>>>(Xsum, cnt, keys);
    som_sqnorm<<<BATCH + G_TOTAL, 256, 0, stream>>>(X, Wm, x2, w2);
    som_dist<<<dim3(BATCH / BM, G_TOTAL / BN), 256, 0, stream>>>(X, Wm, x2, w2, keys);
    som_bmu<<<BATCH / 256, 256, 0, stream>>>(keys, bmu, cnt, out);
    som_scatter<<<BATCH, 256, 0, stream>>>(X, bmu, Xsum);
    som_trans1<<<dim3(64, DIM / 128), 128, 0, stream>>>(Xsum, cnt, tmp, Scol);
    som_update<<<G_TOTAL, 256, 0, stream>>>(Wm, tmp, Scol, out + 3 * BATCH);
}